// LocalNN2d_60756607369257
// MI455X (gfx1250) — compile-verified
//
#include <hip/hip_runtime.h>
#include <math.h>

// ---------------------------------------------------------------------------
// LocalNN2d on MI455X (gfx1250, wave32).
//
// One wave processes TWO 16-row tiles (32 rows) per iteration:
//   * all hidden matmuls on V_WMMA_F32_16X16X4_F32 (full f32, K=4 chunks,
//     bias folded into the C accumulator),
//   * LayerNorm + exact GELU in a row-per-lane layout (lane l owns row l of
//     the 32-row group): tree-reduced in-register stats (depth 4, 4-way ILP),
//     branchless A&S erf, no cross-lane ops,
//   * D-layout <-> row layout conversions through a per-wave LDS tile with
//     row stride 20 floats (bank-conflict-free on 64 banks for the
//     column-per-lane D store, the half-wave-phased b128 row accesses, and
//     the b64 A-layout gather),
//   * 16->4 output head as per-lane dots with Wo/bo/gamma/beta read through
//     uniform scalar loads (SGPR operands), 1+tanh as 2*rcp(1+exp2(-2log2e z)),
//     one coalesced non-temporal b128 store per lane.
// ---------------------------------------------------------------------------

typedef __attribute__((ext_vector_type(2))) float v2f;
typedef __attribute__((ext_vector_type(4))) float v4f;
typedef __attribute__((ext_vector_type(8))) float v8f;

#define N_FEAT   8
#define NN       16
#define N_ENERGY 4
#define LSTRIDE  20   // padded LDS row stride in floats
#define WAVES_PER_BLOCK 8
#define BLOCK_THREADS   256

__device__ __forceinline__ v8f wmma_f32_k4(v2f a, v2f b, v8f c) {
  // (neg_a, A, neg_b, B, c_mod, C, reuse_a, reuse_b)
  return __builtin_amdgcn_wmma_f32_16x16x4_f32(false, a, false, b,
                                               (short)0, c, false, false);
}

// Wave-internal LDS phase boundary: DS ops are in-order per wave; this is a
// compiler barrier + dscnt wait so reg<->LDS round trips stay ordered.
__device__ __forceinline__ void lds_fence() {
  asm volatile("s_wait_dscnt 0" ::: "memory");
}

// Branchless erf: Abramowitz-Stegun 7.1.26, |err| <= 1.5e-7 (f32-exact).
__device__ __forceinline__ float fast_erf(float x) {
  const float ax = fabsf(x);
  const float t  = __builtin_amdgcn_rcpf(fmaf(0.3275911f, ax, 1.0f));
  float p = fmaf(t, 1.061405429f, -1.453152027f);
  p = fmaf(t, p, 1.421413741f);
  p = fmaf(t, p, -0.284496736f);
  p = fmaf(t, p, 0.254829592f);
  p = p * t;
  const float e = __builtin_amdgcn_exp2f(-ax * ax * 1.4426950408889634f);
  return copysignf(fmaf(-p, e, 1.0f), x);
}

__device__ __forceinline__ float gelu_exact(float h) {
  return 0.5f * h * (1.0f + fast_erf(h * 0.70710678118654752f));
}

// Per-lane LayerNorm over the 16 features of this lane's row, then GELU.
// Stats are tree-reduced (depth 4) with 4 partial accumulators for ILP on
// the dual-issue VALU. g/be are uniform (scalar-loaded) per feature.
__device__ __forceinline__ void ln_gelu_row(float x[NN],
                                            const float* __restrict__ g,
                                            const float* __restrict__ be) {
  // pairwise tree sum: depth 4 instead of a 15-deep serial chain
  float s01[8];
#pragma unroll
  for (int k = 0; k < 8; ++k) s01[k] = x[2 * k] + x[2 * k + 1];
  float s0 = (s01[0] + s01[1]) + (s01[2] + s01[3]);
  float s1 = (s01[4] + s01[5]) + (s01[6] + s01[7]);
  const float mu = (s0 + s1) * 0.0625f;

  float d[NN];
  float v0 = 0.0f, v1 = 0.0f, v2 = 0.0f, v3 = 0.0f;
#pragma unroll
  for (int k = 0; k < NN; k += 4) {
    d[k + 0] = x[k + 0] - mu;  v0 = fmaf(d[k + 0], d[k + 0], v0);
    d[k + 1] = x[k + 1] - mu;  v1 = fmaf(d[k + 1], d[k + 1], v1);
    d[k + 2] = x[k + 2] - mu;  v2 = fmaf(d[k + 2], d[k + 2], v2);
    d[k + 3] = x[k + 3] - mu;  v3 = fmaf(d[k + 3], d[k + 3], v3);
  }
  const float var  = (v0 + v1) + (v2 + v3);
  const float rstd = __builtin_amdgcn_rsqf(fmaf(var, 0.0625f, 1e-5f));
#pragma unroll
  for (int k = 0; k < NN; ++k) {
    const float h = fmaf(d[k] * rstd, g[k], be[k]);
    x[k] = gelu_exact(h);
  }
}

// One K-chunk of a row-major (K x 16) weight matrix in WMMA B layout:
//   VGPR0: lanes 0-15 = W[4c+0][n], lanes 16-31 = W[4c+2][n]
//   VGPR1: lanes 0-15 = W[4c+1][n], lanes 16-31 = W[4c+3][n]
__device__ __forceinline__ v2f load_bchunk(const float* W, int c, int n, int hb) {
  const int rb = 4 * c + 2 * hb;
  v2f b;
  b.x = W[rb * NN + n];
  b.y = W[(rb + 1) * NN + n];
  return b;
}

// Spill a WMMA C/D tile (16x16, tile row base `base`) to LDS row-major.
__device__ __forceinline__ void store_d(float* t, int base, int n, int hb, const v8f& d) {
#pragma unroll
  for (int r = 0; r < 8; ++r) t[(base + 8 * hb + r) * LSTRIDE + n] = d[r];
}

// Gather the A operand (four 16x4 K-chunks) for one 16-row tile from LDS.
__device__ __forceinline__ void gather_a(const float* t, int base, int n, int hb, v2f a[4]) {
#pragma unroll
  for (int c = 0; c < 4; ++c)
    a[c] = *(const v2f*)(t + (base + n) * LSTRIDE + 4 * c + 2 * hb);
}

__global__ __launch_bounds__(BLOCK_THREADS)
void localnn2d_wmma_kernel(const float* __restrict__ feat,
                           const float* __restrict__ W0, const float* __restrict__ b0,
                           const float* __restrict__ g0, const float* __restrict__ be0,
                           const float* __restrict__ W1, const float* __restrict__ b1,
                           const float* __restrict__ g1, const float* __restrict__ be1,
                           const float* __restrict__ W2, const float* __restrict__ b2,
                           const float* __restrict__ g2, const float* __restrict__ be2,
                           const float* __restrict__ Wo, const float* __restrict__ bo,
                           float* __restrict__ out, int npairs) {
  __shared__ float lds[WAVES_PER_BLOCK][2 * NN * LSTRIDE];  // 32 rows/wave

  const int lane = threadIdx.x & 31;
  const int wid  = threadIdx.x >> 5;
  const int n    = lane & 15;  // WMMA N index / row-within-tile
  const int hb   = lane >> 4;  // half-wave select
  float* t = lds[wid];

  // ---- per-wave preload of lane-dependent parameters ----
  v2f w0c[2], w1c[4], w2c[4];
#pragma unroll
  for (int c = 0; c < 2; ++c) w0c[c] = load_bchunk(W0, c, n, hb);
#pragma unroll
  for (int c = 0; c < 4; ++c) w1c[c] = load_bchunk(W1, c, n, hb);
#pragma unroll
  for (int c = 0; c < 4; ++c) w2c[c] = load_bchunk(W2, c, n, hb);
  const float b0n = b0[n], b1n = b1[n], b2n = b2[n];

  const int gwave  = blockIdx.x * WAVES_PER_BLOCK + wid;
  const int nwaves = gridDim.x * WAVES_PER_BLOCK;

  for (int pair = gwave; pair < npairs; pair += nwaves) {
    const long long rbase = (long long)pair * 32;   // first of 32 rows
    const float* xp = feat + rbase * N_FEAT;

    if (pair + nwaves < npairs)  // uniform predicate, no divergence
      __builtin_prefetch(feat + (long long)(pair + nwaves) * 32 * N_FEAT + lane * 8, 0, 0);

    // ---- layer 0: A operands straight from global (streaming), 2 tiles ----
    // chunk c, lane<16: {X[n][4c], X[n][4c+1]}; lane>=16: {X[n][4c+2], X[n][4c+3]}
    v8f c0, c1;
#pragma unroll
    for (int r = 0; r < 8; ++r) { c0[r] = b0n; c1[r] = b0n; }
#pragma unroll
    for (int T = 0; T < 2; ++T) {
      const float* xt = xp + T * 16 * N_FEAT;
      v2f a0 = __builtin_nontemporal_load((const v2f*)(xt + n * N_FEAT + 0 + 2 * hb));
      v2f a1 = __builtin_nontemporal_load((const v2f*)(xt + n * N_FEAT + 4 + 2 * hb));
      v8f& c = T ? c1 : c0;
      c = wmma_f32_k4(a0, w0c[0], c);
      c = wmma_f32_k4(a1, w0c[1], c);
    }

    float x[NN];  // this lane's row (row index = lane) during LN/GELU phases
#pragma unroll
    for (int L = 0; L < 3; ++L) {
      // spill both D tiles row-major into LDS
      store_d(t, 0, n, hb, c0);
      store_d(t, NN, n, hb, c1);
      lds_fence();

      // row-per-lane LayerNorm + GELU (all 32 lanes busy, no cross-lane ops)
      float* row = t + lane * LSTRIDE;  // 80B stride -> 16B aligned
#pragma unroll
      for (int q = 0; q < 4; ++q) {
        const v4f xv = *(const v4f*)(row + 4 * q);
#pragma unroll
        for (int i = 0; i < 4; ++i) x[4 * q + i] = xv[i];
      }
      if (L == 0)      ln_gelu_row(x, g0, be0);
      else if (L == 1) ln_gelu_row(x, g1, be1);
      else             ln_gelu_row(x, g2, be2);

      if (L == 2) break;  // layer-2 activations stay in registers for the head

      // store activated rows back and regather WMMA A operands
#pragma unroll
      for (int q = 0; q < 4; ++q) {
        v4f xv;
#pragma unroll
        for (int i = 0; i < 4; ++i) xv[i] = x[4 * q + i];
        *(v4f*)(row + 4 * q) = xv;
      }
      lds_fence();
      v2f a0[4], a1[4];
      gather_a(t, 0, n, hb, a0);
      gather_a(t, NN, n, hb, a1);
      lds_fence();

      const v2f* wc = (L == 0) ? w1c : w2c;
      const float bn = (L == 0) ? b1n : b2n;
#pragma unroll
      for (int r = 0; r < 8; ++r) { c0[r] = bn; c1[r] = bn; }
#pragma unroll
      for (int cc = 0; cc < 4; ++cc) c0 = wmma_f32_k4(a0[cc], wc[cc], c0);
#pragma unroll
      for (int cc = 0; cc < 4; ++cc) c1 = wmma_f32_k4(a1[cc], wc[cc], c1);
    }

    // ---- output head: per-lane dot with Wo (uniform -> scalar operands) ----
    float z[N_ENERGY];
#pragma unroll
    for (int j = 0; j < N_ENERGY; ++j) z[j] = bo[j];
#pragma unroll
    for (int k = 0; k < NN; ++k) {
      const float xv = x[k];
#pragma unroll
      for (int j = 0; j < N_ENERGY; ++j) z[j] = fmaf(xv, Wo[k * N_ENERGY + j], z[j]);
    }
    v4f o;
#pragma unroll
    for (int j = 0; j < N_ENERGY; ++j)  // 1 + tanh(z) = 2 / (1 + exp(-2z))
      o[j] = 2.0f * __builtin_amdgcn_rcpf(
                 1.0f + __builtin_amdgcn_exp2f(z[j] * -2.8853900817779268f));
    __builtin_nontemporal_store(o, (v4f*)(out + (rbase + lane) * N_ENERGY));
    lds_fence();  // LDS row reads retired before next iteration's D stores
  }
}

extern "C" void kernel_launch(void* const* d_in, const int* in_sizes, int n_in,
                              void* d_out, int out_size, void* d_ws, size_t ws_size,
                              hipStream_t stream) {
  const float* feat = (const float*)d_in[0];
  const float* W0 = (const float*)d_in[1];
  const float* b0 = (const float*)d_in[2];
  const float* g0 = (const float*)d_in[3];
  const float* be0 = (const float*)d_in[4];
  const float* W1 = (const float*)d_in[5];
  const float* b1 = (const float*)d_in[6];
  const float* g1 = (const float*)d_in[7];
  const float* be1 = (const float*)d_in[8];
  const float* W2 = (const float*)d_in[9];
  const float* b2 = (const float*)d_in[10];
  const float* g2 = (const float*)d_in[11];
  const float* be2 = (const float*)d_in[12];
  const float* Wo = (const float*)d_in[13];
  const float* bo = (const float*)d_in[14];
  float* out = (float*)d_out;

  const long long M = (long long)in_sizes[0] / N_FEAT;  // 4,194,304 rows
  const int npairs = (int)(M / 32);                     // 131,072 tile pairs

  int blocks = 2048;  // 16,384 waves -> 8 tile pairs per wave
  const int max_blocks = (npairs + WAVES_PER_BLOCK - 1) / WAVES_PER_BLOCK;
  if (blocks > max_blocks) blocks = max_blocks;

  localnn2d_wmma_kernel<<<blocks, BLOCK_THREADS, 0, stream>>>(
      feat, W0, b0, g0, be0, W1, b1, g1, be1, W2, b2, g2, be2, Wo, bo,
      out, npairs);
}